// DifferentiableSampler_65549790871626
// MI455X (gfx1250) — compile-verified
//
#include <hip/hip_runtime.h>
#include <hip/hip_bf16.h>
#include <stdint.h>

// Problem constants (match reference)
#define BSEG 16
#define PSEG 125000
#define NPTS (BSEG * PSEG)      // 2,000,000
#define CIN  32
#define HID  64
#define KSEL 62500              // P * 0.5
#define TILES 8                 // 16-row tiles per wave in the MLP kernel
// Radix-sort tiling
#define VT     6
#define CHUNK  (1024 * VT)      // 6144
#define NCHUNK 21               // ceil(125000 / 6144)
#define PPAD   (NCHUNK * CHUNK) // 129024 padded segment length
#define NPAD   (BSEG * PPAD)    // 2,064,384

typedef __attribute__((ext_vector_type(2))) float v2f;
typedef __attribute__((ext_vector_type(8))) float v8f;

// ---------------------------------------------------------------------------
// Kernel 1: per-point MLP logit via V_WMMA_F32_16X16X4_F32, fused with the
// gumbel perturbation and conversion to a sortable inverted-order uint key.
// W1 fragments / b1 / W2 are loaded ONCE per wave (register resident) and
// reused across TILES row-tiles; x streaming sets the roofline (~11 us).
// A-frag (16x4 f32): lanes 0-15 hold K=0,1; lanes 16-31 hold K=2,3 (per ISA).
// C/D layout: VGPR r, lanes 0-15 -> M=r, N=lane; lanes 16-31 -> M=r+8.
// ---------------------------------------------------------------------------
__global__ __launch_bounds__(256)
void mlp_keys_kernel(const float* __restrict__ x,
                     const float* __restrict__ W1, const float* __restrict__ b1,
                     const float* __restrict__ W2, const float* __restrict__ b2,
                     const float* __restrict__ gumbel,
                     unsigned* __restrict__ keys, unsigned* __restrict__ idx)
{
    const int wid   = (blockIdx.x * blockDim.x + threadIdx.x) >> 5;
    const int lane  = threadIdx.x & 31;
    const int mrow  = lane & 15;          // row (A) / column (B,C) within tile
    const int khalf = (lane >> 4) << 1;   // 0 or 2: which K-pair this half-wave owns

    // ---- register-resident B fragments + second-layer params (loaded once)
    v2f   bfrag[4][8];
    float bias[4], w2c[4];
#pragma unroll
    for (int t = 0; t < 4; ++t) {
        const int col = 16 * t + mrow;
#pragma unroll
        for (int k = 0; k < 8; ++k) {
            bfrag[t][k].x = W1[(4 * k + khalf)     * HID + col];
            bfrag[t][k].y = W1[(4 * k + khalf + 1) * HID + col];
        }
        bias[t] = b1[col];
        w2c [t] = W2[col];
    }
    const float bo = b2[0];

    for (int tt = 0; tt < TILES; ++tt) {
        const int row0 = (wid * TILES + tt) * 16;
        if (row0 >= NPTS) break;

        // A fragments: x tile, 8 K-steps of 4 columns each (C = 32)
        const float* xr = x + (size_t)(row0 + mrow) * CIN + khalf;
        v2f a[8];
#pragma unroll
        for (int k = 0; k < 8; ++k) { a[k].x = xr[4 * k]; a[k].y = xr[4 * k + 1]; }
        if (row0 + 16 < NPTS) __builtin_prefetch(xr + 16 * CIN); // next tile -> global_prefetch_b8

        float ssum[8] = {0.f, 0.f, 0.f, 0.f, 0.f, 0.f, 0.f, 0.f};
#pragma unroll
        for (int t = 0; t < 4; ++t) {      // 4 column tiles of H = 64
            v8f c = {};
#pragma unroll
            for (int k = 0; k < 8; ++k)
                c = __builtin_amdgcn_wmma_f32_16x16x4_f32(
                        false, a[k], false, bfrag[t][k], (short)0, c, false, false);
#pragma unroll
            for (int r = 0; r < 8; ++r) {
                float h = c[r] + bias[t];
                h = fmaxf(h, 0.0f);        // ReLU
                ssum[r] += h * w2c[t];     // fold 64->1 layer per column
            }
        }
        // reduce over the 16 lanes holding the same rows (butterfly stays in half)
#pragma unroll
        for (int r = 0; r < 8; ++r) {
            ssum[r] += __shfl_xor(ssum[r], 1);
            ssum[r] += __shfl_xor(ssum[r], 2);
            ssum[r] += __shfl_xor(ssum[r], 4);
            ssum[r] += __shfl_xor(ssum[r], 8);
        }
        if ((lane & 15) == 0) {            // lane 0 -> rows 0..7, lane 16 -> rows 8..15
            const int rbase = row0 + ((lane >> 4) << 3);
#pragma unroll
            for (int r = 0; r < 8; ++r) {
                const int   p    = rbase + r;
                const float keyf = ssum[r] + bo + gumbel[p]; // monotone in y_soft (row-wise)
                unsigned u = __float_as_uint(keyf);
                u ^= (u & 0x80000000u) ? 0xFFFFFFFFu : 0x80000000u; // total order, ascending
                u = ~u;                                             // invert -> descending
                const int s  = p / PSEG;
                const int pl = p - s * PSEG;
                keys[s * PPAD + pl] = u;
                idx [s * PPAD + pl] = (unsigned)p;  // already the *global* index
            }
        }
    }
}

// Fill the per-segment padding with max keys (stable sort keeps them at tail)
__global__ void pad_kernel(unsigned* __restrict__ keys, unsigned* __restrict__ idx)
{
    const int padPer = PPAD - PSEG;  // 4024
    int t = blockIdx.x * blockDim.x + threadIdx.x;
    if (t >= BSEG * padPer) return;
    int s = t / padPer;
    int o = t - s * padPer;
    keys[s * PPAD + PSEG + o] = 0xFFFFFFFFu;
    idx [s * PPAD + PSEG + o] = 0u;
}

// ---------------------------------------------------------------------------
// Kernel 2: one stable LSD radix-256 pass. One 1024-thread WG per segment,
// 6 items/thread. Binary splits use a two-level wave32 shuffle scan
// (3 barriers/split instead of ~22 with an LDS Hillis-Steele scan).
// Stable => equal keys keep ascending-index order (top_k tie-break).
// ---------------------------------------------------------------------------
__global__ __launch_bounds__(1024)
void radix_pass_kernel(const unsigned* __restrict__ kin, const unsigned* __restrict__ iin,
                       unsigned* __restrict__ kout, unsigned* __restrict__ iout,
                       int shift)
{
    __shared__ unsigned lkey[CHUNK];
    __shared__ unsigned lidx[CHUNK];
    __shared__ unsigned swtot[32];
    __shared__ unsigned totz;
    __shared__ unsigned cursor[256];
    __shared__ unsigned chist[256];
    __shared__ unsigned dstart[256];

    const int tid  = threadIdx.x;
    const int lane = tid & 31;
    const int wv   = tid >> 5;
    const int seg  = blockIdx.x;
    const unsigned base = (unsigned)seg * PPAD;

    // ---- Phase 1: segment-wide digit histogram -> cursor (exclusive scan)
    if (tid < 256) chist[tid] = 0u;
    __syncthreads();
    for (int i = tid * 4; i < PPAD; i += 4096) {
        uint4 kk = *(const uint4*)(kin + base + i);
        atomicAdd(&chist[(kk.x >> shift) & 255u], 1u);
        atomicAdd(&chist[(kk.y >> shift) & 255u], 1u);
        atomicAdd(&chist[(kk.z >> shift) & 255u], 1u);
        atomicAdd(&chist[(kk.w >> shift) & 255u], 1u);
    }
    __syncthreads();
    if (tid == 0) {
        unsigned run = 0;
        for (int d = 0; d < 256; ++d) { unsigned c = chist[d]; cursor[d] = run; run += c; }
    }
    __syncthreads();

    // ---- Phase 2: sequential chunks, stable local sort + stable scatter
    for (int ch = 0; ch < NCHUNK; ++ch) {
        const unsigned cbase = base + (unsigned)ch * CHUNK;
        unsigned k[VT], id[VT];
#pragma unroll
        for (int j = 0; j < VT; ++j) {
            k[j]  = kin[cbase + VT * tid + j];
            id[j] = iin[cbase + VT * tid + j];
        }
        // 8 stable binary splits on the current digit's bits
        for (int b = 0; b < 8; ++b) {
            unsigned bit[VT];
            unsigned myZeros = 0;
#pragma unroll
            for (int j = 0; j < VT; ++j) {
                bit[j] = (k[j] >> (shift + b)) & 1u;
                myZeros += bit[j] ^ 1u;
            }
            // wave-level inclusive scan of per-thread zero counts
            unsigned incl = myZeros;
#pragma unroll
            for (int off = 1; off < 32; off <<= 1) {
                unsigned v = __shfl_up(incl, off);
                if (lane >= off) incl += v;
            }
            if (lane == 31) swtot[wv] = incl;
            __syncthreads();
            if (tid < 32) {                 // wave 0 scans the 32 wave totals
                unsigned v = swtot[tid], iv = v;
#pragma unroll
                for (int off = 1; off < 32; off <<= 1) {
                    unsigned u = __shfl_up(iv, off);
                    if (tid >= off) iv += u;
                }
                swtot[tid] = iv - v;        // exclusive wave prefix
                if (tid == 31) totz = iv;   // total zeros in chunk
            }
            __syncthreads();
            const unsigned inclAll     = incl + swtot[wv];
            const unsigned total       = totz;
            const unsigned zerosBefore = inclAll - myZeros;
            const unsigned onesBefore  = (unsigned)(VT * tid) - zerosBefore;
            unsigned z = 0, o = 0;
#pragma unroll
            for (int j = 0; j < VT; ++j) {
                unsigned pos = bit[j] ? (total + onesBefore + o++)
                                      : (zerosBefore + z++);
                lkey[pos] = k[j];
                lidx[pos] = id[j];
            }
            __syncthreads();
#pragma unroll
            for (int j = 0; j < VT; ++j) {
                k[j]  = lkey[VT * tid + j];
                id[j] = lidx[VT * tid + j];
            }
            // next split's first barrier protects lkey/lidx reuse
        }
        __syncthreads();
        // chunk digit counts + digit start positions (chunk now digit-sorted)
        if (tid < 256) chist[tid] = 0u;
        __syncthreads();
#pragma unroll
        for (int j = 0; j < VT; ++j) {
            const unsigned q   = (unsigned)(VT * tid) + j;
            const unsigned dig = (k[j] >> shift) & 255u;
            atomicAdd(&chist[dig], 1u);
            unsigned prevdig = 0xFFFFFFFFu;
            if (q > 0) prevdig = (lkey[q - 1] >> shift) & 255u;
            if (dig != prevdig) dstart[dig] = q;
        }
        __syncthreads();
#pragma unroll
        for (int j = 0; j < VT; ++j) {
            const unsigned q    = (unsigned)(VT * tid) + j;
            const unsigned dig  = (k[j] >> shift) & 255u;
            const unsigned gpos = cursor[dig] + (q - dstart[dig]);
            kout[base + gpos] = k[j];
            iout[base + gpos] = id[j];
        }
        __syncthreads();
        if (tid < 256) cursor[tid] += chist[tid];
        __syncthreads();
    }
}

// First K sorted indices per segment -> output (int32 written through d_out)
__global__ void gather_out_kernel(const unsigned* __restrict__ idx, int* __restrict__ out)
{
    int t = blockIdx.x * blockDim.x + threadIdx.x;
    if (t >= BSEG * KSEL) return;
    int s = t / KSEL;
    int o = t - s * KSEL;
    out[t] = (int)idx[s * PPAD + o];
}

extern "C" void kernel_launch(void* const* d_in, const int* in_sizes, int n_in,
                              void* d_out, int out_size, void* d_ws, size_t ws_size,
                              hipStream_t stream)
{
    const float* x   = (const float*)d_in[0];
    // d_in[1] = batch (int64): unused — segments are contiguous equal blocks
    const float* W1  = (const float*)d_in[2];
    const float* b1  = (const float*)d_in[3];
    const float* W2  = (const float*)d_in[4];
    const float* b2  = (const float*)d_in[5];
    const float* gum = (const float*)d_in[6];

    unsigned* ws    = (unsigned*)d_ws;     // needs 4 * NPAD * 4 B ~= 33 MB
    unsigned* keysA = ws;
    unsigned* idxA  = ws + (size_t)NPAD;
    unsigned* keysB = ws + 2 * (size_t)NPAD;
    unsigned* idxB  = ws + 3 * (size_t)NPAD;

    pad_kernel<<<(BSEG * (PPAD - PSEG) + 255) / 256, 256, 0, stream>>>(keysA, idxA);

    // 125000 tiles / 8 per wave = 15625 waves; 8 waves per block -> 1954 blocks
    const int nwaves  = (NPTS / 16 + TILES - 1) / TILES;
    const int nblocks = (nwaves + 7) / 8;
    mlp_keys_kernel<<<nblocks, 256, 0, stream>>>(x, W1, b1, W2, b2, gum, keysA, idxA);

    radix_pass_kernel<<<BSEG, 1024, 0, stream>>>(keysA, idxA, keysB, idxB, 0);
    radix_pass_kernel<<<BSEG, 1024, 0, stream>>>(keysB, idxB, keysA, idxA, 8);
    radix_pass_kernel<<<BSEG, 1024, 0, stream>>>(keysA, idxA, keysB, idxB, 16);
    radix_pass_kernel<<<BSEG, 1024, 0, stream>>>(keysB, idxB, keysA, idxA, 24);

    gather_out_kernel<<<(BSEG * KSEL + 255) / 256, 256, 0, stream>>>(idxA, (int*)d_out);
}